// MetalKernelPagedAttentionWrapper_69106023793392
// MI455X (gfx1250) — compile-verified
//
#include <hip/hip_runtime.h>
#include <hip/hip_bf16.h>

// ---------------- constants (match reference) ----------------
#define BATCH     32
#define DMODEL    4096
#define NHEADS    32
#define NKV       8
#define GQA       4          // NHEADS / NKV
#define HEADDIM   128
#define HALFDIM   64
#define BLKSZ     16
#define MAXSEQ    2048
#define MAXBLK    128        // MAXSEQ / BLKSZ
#define SCALE_F   0.08838834764831845f   // 128^-0.5
#define ROPE_BASE 10000.0f

typedef __attribute__((ext_vector_type(2))) float v2f;
typedef __attribute__((ext_vector_type(4))) float v4f;
typedef __attribute__((ext_vector_type(8))) float v8f;

static __device__ __forceinline__ v8f wmma4(const v2f a, const v2f b, v8f c) {
    return __builtin_amdgcn_wmma_f32_16x16x4_f32(
        false, a, false, b, (short)0, c, false, false);
}

// ---------------------------------------------------------------------------
// GEMM: P[split][32 x N] = X[32 x Kc] @ W[Kc x N] (partial over K chunk).
// Compile-time K/N so strided weight accesses fold into 24-bit immediates.
// Block = 128 threads = 4 waves. Wave tile = 32(M) x 32(N): 4 independent
// V_WMMA_F32_16X16X4_F32 per K-step; every weight row is streamed exactly
// once from HBM (non-temporal). grid = (N/128, SPLITK).
// ---------------------------------------------------------------------------
template<int K, int N, int SPLITK>
__global__ __launch_bounds__(128) void gemm_wmma_f32(
    const float* __restrict__ X, const float* __restrict__ W,
    float* __restrict__ P)
{
    constexpr int KCHUNK = K / SPLITK;

    const int lane = threadIdx.x & 31;
    const int wave = threadIdx.x >> 5;
    const int n0   = (blockIdx.x * 4 + wave) * 32;
    const int ks   = blockIdx.y * KCHUNK;

    const int l16  = lane & 15;
    const int half = lane >> 4;          // 0: K pair {0,1} | 1: K pair {2,3}
    const int kA   = half * 2;

    const float* Ap0 = X + (size_t)l16 * K + ks + kA;      // M rows 0..15
    const float* Ap1 = Ap0 + (size_t)16 * K;               // M rows 16..31
    const float* Bp  = W + (size_t)(ks + kA) * N + (n0 + l16);

    v8f acc00 = {0.f,0.f,0.f,0.f,0.f,0.f,0.f,0.f};   // M 0-15 , N +0
    v8f acc01 = acc00;                               // M 0-15 , N +16
    v8f acc10 = acc00;                               // M 16-31, N +0
    v8f acc11 = acc00;                               // M 16-31, N +16

    #pragma unroll 8
    for (int k0 = 0; k0 < KCHUNK; k0 += 4) {
        if ((k0 & 127) == 0 && (k0 + 512) < KCHUNK) {
            __builtin_prefetch(Bp + (size_t)(k0 + 512) * N, 0, 0);
        }
        const v2f a0 = *reinterpret_cast<const v2f*>(Ap0 + k0);
        const v2f a1 = *reinterpret_cast<const v2f*>(Ap1 + k0);
        v2f b0, b1;
        b0[0] = __builtin_nontemporal_load(Bp + (size_t)(k0    ) * N);
        b0[1] = __builtin_nontemporal_load(Bp + (size_t)(k0 + 1) * N);
        b1[0] = __builtin_nontemporal_load(Bp + (size_t)(k0    ) * N + 16);
        b1[1] = __builtin_nontemporal_load(Bp + (size_t)(k0 + 1) * N + 16);
        acc00 = wmma4(a0, b0, acc00);
        acc10 = wmma4(a1, b0, acc10);
        acc01 = wmma4(a0, b1, acc01);
        acc11 = wmma4(a1, b1, acc11);
    }

    // C/D layout: VGPR r -> row (half*8 + r), col l16
    float* P0 = P + ((size_t)blockIdx.y * BATCH + half * 8) * N + (n0 + l16);
    float* P1 = P0 + (size_t)16 * N;
    #pragma unroll
    for (int r = 0; r < 8; ++r) {
        P0[(size_t)r * N]      = acc00[r];
        P0[(size_t)r * N + 16] = acc01[r];
        P1[(size_t)r * N]      = acc10[r];
        P1[(size_t)r * N + 16] = acc11[r];
    }
}

// Y[i] = P[i] + P[i+n]  (fold the two split-K partials)
__global__ __launch_bounds__(256) void reduce_split2(
    const float* __restrict__ P, float* __restrict__ Y, int n)
{
    const int i = blockIdx.x * 256 + threadIdx.x;
    if (i < n) Y[i] = P[i] + P[i + n];
}

// ---------------------------------------------------------------------------
// RoPE (in place) on q [B,32,128] and k [B,8,128].
// grid = (B, NHEADS + NKV), block = 64 (one thread per rotation pair).
// ---------------------------------------------------------------------------
__global__ __launch_bounds__(64) void rope_kernel(
    float* __restrict__ q, float* __restrict__ k,
    const int* __restrict__ ctx)
{
    const int b = blockIdx.x;
    const int h = blockIdx.y;            // 0..39: 0-31 q heads, 32-39 k heads
    const int i = threadIdx.x;           // 0..63
    const int pos = ctx[b] - 1;

    const float inv_freq = powf(ROPE_BASE, -(float)i / (float)HALFDIM);
    const float ang = (float)pos * inv_freq;
    float s, c;
    sincosf(ang, &s, &c);

    float* v = (h < NHEADS)
        ? (q + ((size_t)b * NHEADS + h) * HEADDIM)
        : (k + ((size_t)b * NKV + (h - NHEADS)) * HEADDIM);

    const float x1 = v[i];
    const float x2 = v[i + HALFDIM];
    v[i]           = x1 * c - x2 * s;
    v[i + HALFDIM] = x1 * s + x2 * c;
}

// ---------------------------------------------------------------------------
// Paged attention, one workgroup (128 threads) per (batch, kv-head).
// Phase 1: scores via 8-lane cooperative dots (16 key rows in flight).
// Phase 2: softmax, one wave32 per query head of the group.
// Phase 3: P @ V with fully-coalesced 512B V-row reads (thread d = dim).
// Fresh K/V for t == L-1 comes from workspace (cache never mutated).
// K/V cache streamed with non-temporal hints (touch-once, 536 MB total).
// ---------------------------------------------------------------------------
__global__ __launch_bounds__(128) void attn_kernel(
    const float* __restrict__ q,            // [B,32,128] (rotated)
    const float* __restrict__ kn,           // [B,8,128]  (rotated new k)
    const float* __restrict__ vn,           // [B,8,128]  (new v)
    const float* __restrict__ key_cache,    // [slots,8,128]
    const float* __restrict__ value_cache,  // [slots,8,128]
    const int*   __restrict__ block_tables, // [B,128]
    const int*   __restrict__ ctx,          // [B]
    float* __restrict__ out)                // [B,32,128]
{
    __shared__ float sq[GQA][HEADDIM];
    __shared__ float sp[GQA][MAXSEQ];
    __shared__ int   sbt[MAXBLK];
    __shared__ float sinv[GQA];

    const int b   = blockIdx.x;
    const int kvh = blockIdx.y;
    const int tid = threadIdx.x;
    const int L   = ctx[b];

    for (int idx = tid; idx < GQA * HEADDIM; idx += 128) {
        const int g = idx >> 7, d = idx & 127;
        sq[g][d] = q[((size_t)b * NHEADS + kvh * GQA + g) * HEADDIM + d];
    }
    sbt[tid] = block_tables[b * MAXBLK + tid];
    __syncthreads();

    const float* knew = kn + ((size_t)b * NKV + kvh) * HEADDIM;
    const float* vnew = vn + ((size_t)b * NKV + kvh) * HEADDIM;

    // ---- phase 1: scores -------------------------------------------------
    {
        const int grp = tid >> 3;   // 16 key rows in flight
        const int l8  = tid & 7;    // 16 dims each
        for (int t = grp; t < L; t += 16) {
            const float* krow;
            if (t == L - 1) {
                krow = knew;
            } else {
                const int slot = sbt[t >> 4] * BLKSZ + (t & 15);
                krow = key_cache + ((size_t)slot * NKV + kvh) * HEADDIM;
            }
            float s0 = 0.f, s1 = 0.f, s2 = 0.f, s3 = 0.f;
            #pragma unroll
            for (int j = 0; j < 4; ++j) {
                const int base = l8 * 16 + j * 4;
                const v4f kk = __builtin_nontemporal_load(
                    reinterpret_cast<const v4f*>(krow + base));
                s0 += sq[0][base]*kk[0] + sq[0][base+1]*kk[1] + sq[0][base+2]*kk[2] + sq[0][base+3]*kk[3];
                s1 += sq[1][base]*kk[0] + sq[1][base+1]*kk[1] + sq[1][base+2]*kk[2] + sq[1][base+3]*kk[3];
                s2 += sq[2][base]*kk[0] + sq[2][base+1]*kk[1] + sq[2][base+2]*kk[2] + sq[2][base+3]*kk[3];
                s3 += sq[3][base]*kk[0] + sq[3][base+1]*kk[1] + sq[3][base+2]*kk[2] + sq[3][base+3]*kk[3];
            }
            #pragma unroll
            for (int off = 4; off > 0; off >>= 1) {
                s0 += __shfl_xor(s0, off);
                s1 += __shfl_xor(s1, off);
                s2 += __shfl_xor(s2, off);
                s3 += __shfl_xor(s3, off);
            }
            if (l8 == 0) {
                sp[0][t] = s0 * SCALE_F;
                sp[1][t] = s1 * SCALE_F;
                sp[2][t] = s2 * SCALE_F;
                sp[3][t] = s3 * SCALE_F;
            }
        }
    }
    __syncthreads();

    // ---- phase 2: softmax (one wave32 per query head) --------------------
    {
        const int g    = tid >> 5;
        const int lane = tid & 31;
        float m = -3.0e38f;
        for (int t = lane; t < L; t += 32) m = fmaxf(m, sp[g][t]);
        #pragma unroll
        for (int off = 16; off > 0; off >>= 1) m = fmaxf(m, __shfl_xor(m, off));
        float sum = 0.f;
        for (int t = lane; t < L; t += 32) {
            const float e = __expf(sp[g][t] - m);
            sp[g][t] = e;
            sum += e;
        }
        #pragma unroll
        for (int off = 16; off > 0; off >>= 1) sum += __shfl_xor(sum, off);
        if (lane == 0) sinv[g] = 1.0f / sum;
    }
    __syncthreads();

    // ---- phase 3: out = P @ V --------------------------------------------
    {
        const int d = tid;   // 0..127
        float a0 = 0.f, a1 = 0.f, a2 = 0.f, a3 = 0.f;
        for (int t = 0; t < L; ++t) {
            const float* vrow;
            if (t == L - 1) {
                vrow = vnew;
            } else {
                const int slot = sbt[t >> 4] * BLKSZ + (t & 15);
                vrow = value_cache + ((size_t)slot * NKV + kvh) * HEADDIM;
            }
            const float v = __builtin_nontemporal_load(vrow + d);
            a0 += sp[0][t] * v;
            a1 += sp[1][t] * v;
            a2 += sp[2][t] * v;
            a3 += sp[3][t] * v;
        }
        const size_t ob = ((size_t)b * NHEADS + kvh * GQA) * HEADDIM + d;
        out[ob]               = a0 * sinv[0];
        out[ob + 1 * HEADDIM] = a1 * sinv[1];
        out[ob + 2 * HEADDIM] = a2 * sinv[2];
        out[ob + 3 * HEADDIM] = a3 * sinv[3];
    }
}

// ---------------------------------------------------------------------------
extern "C" void kernel_launch(void* const* d_in, const int* in_sizes, int n_in,
                              void* d_out, int out_size, void* d_ws, size_t ws_size,
                              hipStream_t stream) {
    const float* x            = (const float*)d_in[0];  // [32,1,4096]
    const float* Wq           = (const float*)d_in[1];  // [4096,4096]
    const float* Wk           = (const float*)d_in[2];  // [4096,1024]
    const float* Wv           = (const float*)d_in[3];  // [4096,1024]
    const float* Wo           = (const float*)d_in[4];  // [4096,4096]
    const float* key_cache    = (const float*)d_in[5];
    const float* value_cache  = (const float*)d_in[6];
    const int*   block_tables = (const int*)d_in[7];
    const int*   ctx          = (const int*)d_in[8];
    float*       out          = (float*)d_out;

    constexpr int QN = NHEADS * HEADDIM;   // 4096
    constexpr int KN = NKV * HEADDIM;      // 1024

    float* ws      = (float*)d_ws;
    float* part_q  = ws;                               // 2 * 32*4096
    float* part_k  = part_q + 2 * BATCH * QN;          // 2 * 32*1024
    float* part_v  = part_k + 2 * BATCH * KN;          // 2 * 32*1024
    float* ws_q    = part_v + 2 * BATCH * KN;          // 32*4096
    float* ws_k    = ws_q + BATCH * QN;                // 32*1024
    float* ws_v    = ws_k + BATCH * KN;                // 32*1024
    float* ws_attn = ws_v + BATCH * KN;                // 32*4096
    float* part_o  = part_q;                           // reuse: part_q dead by then

    // QKV projections (fp32 WMMA, split-K=2)
    gemm_wmma_f32<DMODEL, QN, 2><<<dim3(QN / 128, 2), 128, 0, stream>>>(x, Wq, part_q);
    gemm_wmma_f32<DMODEL, KN, 2><<<dim3(KN / 128, 2), 128, 0, stream>>>(x, Wk, part_k);
    gemm_wmma_f32<DMODEL, KN, 2><<<dim3(KN / 128, 2), 128, 0, stream>>>(x, Wv, part_v);

    reduce_split2<<<(BATCH * QN + 255) / 256, 256, 0, stream>>>(part_q, ws_q, BATCH * QN);
    reduce_split2<<<(BATCH * KN + 255) / 256, 256, 0, stream>>>(part_k, ws_k, BATCH * KN);
    reduce_split2<<<(BATCH * KN + 255) / 256, 256, 0, stream>>>(part_v, ws_v, BATCH * KN);

    // RoPE on q and k
    rope_kernel<<<dim3(BATCH, NHEADS + NKV), 64, 0, stream>>>(ws_q, ws_k, ctx);

    // Paged attention (bandwidth-bound streaming of K/V cache)
    attn_kernel<<<dim3(BATCH, NKV), 128, 0, stream>>>(
        ws_q, ws_k, ws_v, key_cache, value_cache, block_tables, ctx, ws_attn);

    // Output projection (fp32 WMMA, split-K=2)
    gemm_wmma_f32<DMODEL, QN, 2><<<dim3(QN / 128, 2), 128, 0, stream>>>(ws_attn, Wo, part_o);
    reduce_split2<<<(BATCH * QN + 255) / 256, 256, 0, stream>>>(part_o, out, BATCH * QN);
}